// VQVAE_52776558133745
// MI455X (gfx1250) — compile-verified
//
#include <hip/hip_runtime.h>
#include <hip/hip_bf16.h>

typedef __attribute__((ext_vector_type(2))) float v2f;
typedef __attribute__((ext_vector_type(8))) float v8f;

#define N_PTS   32768      // B*H*W = 8*64*64
#define K_CODES 8192
#define KCHUNK  2048       // codes staged in LDS per pass (40KB LDS)
#define NSTAGES (K_CODES / KCHUNK)
#define NCH     (KCHUNK / 16)             // 128 WMMA chunks per stage
#define WAVES_PER_BLOCK 8
#define THREADS (WAVES_PER_BLOCK * 32)
#define NTILES  (N_PTS / 16)              // 2048 point tiles
#define NBLOCKS (NTILES / WAVES_PER_BLOCK) // 256

// ---------------------------------------------------------------------------
// Kernel 1: 1x1 "pre-quant conv": z[n][o] = sum_c z_e_in[b][c][hw] * pq_w[o][c] + pq_b[o]
// Writes z as [N][4] contiguous for WMMA A-tile loads.
// ---------------------------------------------------------------------------
__global__ __launch_bounds__(256) void vq_prep(const float* __restrict__ z_e_in,
                                               const float* __restrict__ pq_w,
                                               const float* __restrict__ pq_b,
                                               float* __restrict__ z) {
    int i = blockIdx.x * 256 + threadIdx.x;
    if (i >= N_PTS) return;
    int b = i >> 12;            // HW = 4096
    int hw = i & 4095;
    const float* in = z_e_in + (b * 4) * 4096 + hw;
    float x0 = in[0];
    float x1 = in[4096];
    float x2 = in[2 * 4096];
    float x3 = in[3 * 4096];
    float4 o;
    o.x = pq_b[0] + x0 * pq_w[0]  + x1 * pq_w[1]  + x2 * pq_w[2]  + x3 * pq_w[3];
    o.y = pq_b[1] + x0 * pq_w[4]  + x1 * pq_w[5]  + x2 * pq_w[6]  + x3 * pq_w[7];
    o.z = pq_b[2] + x0 * pq_w[8]  + x1 * pq_w[9]  + x2 * pq_w[10] + x3 * pq_w[11];
    o.w = pq_b[3] + x0 * pq_w[12] + x1 * pq_w[13] + x2 * pq_w[14] + x3 * pq_w[15];
    ((float4*)z)[i] = o;
}

// ---------------------------------------------------------------------------
// Kernel 2: WMMA nearest-code search + gather + per-wave loss partials.
// Each wave owns 16 points; scores = z.cb^T with -0.5*||c||^2 folded into C.
// Inner loop handles TWO 16-code chunks per iteration: both chunks' LDS loads
// issue up-front so chunk A computes under a partial dscnt wait while chunk
// B's loads are in flight; chunk B's WMMA+VALU covers chunk A's hazards.
// ---------------------------------------------------------------------------
__global__ __launch_bounds__(THREADS) void vq_main(const float* __restrict__ z,
                                                   const float* __restrict__ codebook,
                                                   float* __restrict__ latent,
                                                   float* __restrict__ partial) {
    __shared__ float s_cb[KCHUNK * 4];   // 32KB: staged codebook rows
    __shared__ float s_bias[KCHUNK];     // 8KB:  -0.5*||c||^2

    const int tid  = threadIdx.x;
    const int lane = tid & 31;
    const int wave = tid >> 5;
    const int tileBase = (blockIdx.x * WAVES_PER_BLOCK + wave) * 16;

    // A tile (loop-invariant): lane l holds z[tile + (l&15)][khalf, khalf+1]
    const int m     = lane & 15;
    const int khalf = (lane >> 4) << 1;   // 0 for lanes 0-15, 2 for lanes 16-31
    v2f a = *(const v2f*)(z + (tileBase + m) * 4 + khalf);

    float best[8];
    int   bidx[8];
#pragma unroll
    for (int r = 0; r < 8; ++r) { best[r] = -3.402823e38f; bidx[r] = 0; }

    for (int stage = 0; stage < NSTAGES; ++stage) {
        const int kbase = stage * KCHUNK;
        // Cooperative stage: 2048 float4 rows by 256 threads (8 rows each),
        // computing the column bias on the fly.  Prefetch next stage (L2-resident).
        for (int i = tid; i < KCHUNK; i += THREADS) {
            float4 row = ((const float4*)codebook)[kbase + i];
            ((float4*)s_cb)[i] = row;
            s_bias[i] = -0.5f * (row.x * row.x + row.y * row.y +
                                 row.z * row.z + row.w * row.w);
            if (stage + 1 < NSTAGES)
                __builtin_prefetch(&((const float4*)codebook)[kbase + KCHUNK + i], 0, 0);
        }
        __syncthreads();

        for (int it = 0; it < NCH; it += 2) {
            const int colA = (it << 4) + m;
            const int colB = colA + 16;
            // issue both chunks' LDS loads back-to-back (DS returns in order)
            v2f   bA    = *(const v2f*)(s_cb + colA * 4 + khalf);
            float biasA = s_bias[colA];
            v2f   bB    = *(const v2f*)(s_cb + colB * 4 + khalf);
            float biasB = s_bias[colB];

            // ---- chunk A ----
            v8f cA = {biasA, biasA, biasA, biasA, biasA, biasA, biasA, biasA};
            v8f dA = __builtin_amdgcn_wmma_f32_16x16x4_f32(
                false, a, false, bA, (short)0, cA, false, false);
            // ---- chunk B (independent; overlaps A's result hazard) ----
            v8f cB = {biasB, biasB, biasB, biasB, biasB, biasB, biasB, biasB};
            v8f dB = __builtin_amdgcn_wmma_f32_16x16x4_f32(
                false, a, false, bB, (short)0, cB, false, false);

            const int kgA = kbase + colA;
            const int kgB = kbase + colB;
#pragma unroll
            for (int r = 0; r < 8; ++r) {
                if (dA[r] > best[r]) { best[r] = dA[r]; bidx[r] = kgA; }
            }
#pragma unroll
            for (int r = 0; r < 8; ++r) {
                if (dB[r] > best[r]) { best[r] = dB[r]; bidx[r] = kgB; }
            }
        }
        __syncthreads();
    }

    // Butterfly reduce across the 16 lanes sharing each row-half (wave32).
#pragma unroll
    for (int r = 0; r < 8; ++r) {
#pragma unroll
        for (int s = 1; s < 16; s <<= 1) {
            float ob = __shfl_xor(best[r], s, 32);
            int   oi = __shfl_xor(bidx[r], s, 32);
            if (ob > best[r] || (ob == best[r] && oi < bidx[r])) {
                best[r] = ob; bidx[r] = oi;
            }
        }
    }

    // Lanes 0 and 16 each own 8 rows: gather codebook row, scatter to NCHW
    // latent, accumulate (z_q - z)^2 partial.
    float lsum = 0.0f;
    if ((lane & 15) == 0) {
        const int rowOff = (lane >> 4) * 8;   // 0 or 8
#pragma unroll
        for (int r = 0; r < 8; ++r) {
            const int n   = tileBase + rowOff + r;
            const int idx = bidx[r];
            float4 cb = ((const float4*)codebook)[idx];
            float4 zv = ((const float4*)z)[n];
            const int b  = n >> 12;
            const int hw = n & 4095;
            float* o = latent + (b * 4) * 4096 + hw;
            o[0]        = cb.x;
            o[4096]     = cb.y;
            o[2 * 4096] = cb.z;
            o[3 * 4096] = cb.w;
            float dx = cb.x - zv.x, dy = cb.y - zv.y;
            float dz = cb.z - zv.z, dw = cb.w - zv.w;
            lsum += dx * dx + dy * dy + dz * dz + dw * dw;
        }
    }
    lsum += __shfl_xor(lsum, 16, 32);   // combine the two half-wave writers
    if (lane == 0)
        partial[blockIdx.x * WAVES_PER_BLOCK + wave] = lsum;
}

// ---------------------------------------------------------------------------
// Kernel 3: deterministic fixed-order reduction of 2048 partials -> q_loss.
// q_loss = (1 + BETA) * mean((zq - z)^2), BETA = 0.25, mean over N*4 elems.
// ---------------------------------------------------------------------------
__global__ __launch_bounds__(256) void vq_loss(const float* __restrict__ partial,
                                               float* __restrict__ out_loss) {
    __shared__ float s[256];
    float v = 0.0f;
    for (int i = threadIdx.x; i < NTILES; i += 256) v += partial[i];
    s[threadIdx.x] = v;
    __syncthreads();
    for (int w = 128; w > 0; w >>= 1) {
        if (threadIdx.x < (unsigned)w) s[threadIdx.x] += s[threadIdx.x + w];
        __syncthreads();
    }
    if (threadIdx.x == 0)
        out_loss[0] = s[0] * (1.25f / (float)(N_PTS * 4));
}

// ---------------------------------------------------------------------------
extern "C" void kernel_launch(void* const* d_in, const int* in_sizes, int n_in,
                              void* d_out, int out_size, void* d_ws, size_t ws_size,
                              hipStream_t stream) {
    const float* z_e_in   = (const float*)d_in[0];  // [8,4,64,64]
    const float* pq_w     = (const float*)d_in[1];  // [4,4]
    const float* pq_b     = (const float*)d_in[2];  // [4]
    const float* codebook = (const float*)d_in[3];  // [8192,4]

    float* latent = (float*)d_out;            // 131072 floats [8,4,64,64]
    float* loss   = latent + N_PTS * 4;       // 1 float (outputs concatenated)

    float* z       = (float*)d_ws;            // [N][4]  = 131072 floats
    float* partial = z + N_PTS * 4;           // [2048]  per-wave loss partials

    vq_prep<<<N_PTS / 256, 256, 0, stream>>>(z_e_in, pq_w, pq_b, z);
    vq_main<<<NBLOCKS, THREADS, 0, stream>>>(z, codebook, latent, partial);
    vq_loss<<<1, 256, 0, stream>>>(partial, loss);
}